// ImageBuffer_86784109183359
// MI455X (gfx1250) — compile-verified
//
#include <hip/hip_runtime.h>
#include <hip/hip_bf16.h>

// ---------------------------------------------------------------------------
// ImageBuffer sliding-window replicate:
//   out[i][k][:] = (j < 16 ? buffer[j] : inputs[j-16]),  j = i+1+k
//   i in [0,128), k in [0,16), frame = 128*128*3 f32 = 49152 floats = 12288 float4
//
// Pure data movement, write-bound: 402.6 MB out / 28.3 MB unique in.
// Strategy (CDNA5): read each source-frame chunk ONCE from global into LDS via
// async DMA (GLOBAL_LOAD_ASYNC_TO_LDS_B128), then fan out to <=16 output
// windows via GLOBAL_STORE_ASYNC_FROM_LDS_B128 with non-temporal policy
// (output > L2, never re-read). Sync with ASYNCcnt only — each thread stores
// only the LDS lines it loaded itself, so no workgroup barrier is needed.
// ---------------------------------------------------------------------------

#define BATCH   128
#define KBUF    16
#define FR4     12288        // float4 per frame (128*128*3/4)
#define CHUNK4  1024         // float4 per block chunk (16 KB LDS)
#define NCHUNK  (FR4 / CHUNK4)   // 12
#define NSRC    (BATCH + KBUF - 1) // 143 used source frames, j = 1..143
#define TPB     256
#define F4_PER_THREAD (CHUNK4 / TPB) // 4

// Native clang vector (usable with __builtin_nontemporal_* and clean b128 codegen)
typedef float vfloat4 __attribute__((ext_vector_type(4)));

typedef __attribute__((address_space(3))) void lds_void_t;

static __device__ __forceinline__ unsigned lds_byte_offset(void* p) {
    // generic -> LDS addrspace (addrspacecast) -> 32-bit LDS byte address
    return (unsigned)(unsigned long long)(lds_void_t*)p;
}

static __device__ __forceinline__ void wait_asynccnt0() {
#if __has_builtin(__builtin_amdgcn_s_wait_asynccnt)
    __builtin_amdgcn_s_wait_asynccnt(0);
#else
    asm volatile("s_wait_asynccnt 0" ::: "memory");
#endif
}

// ------------------------- async LDS fan-out kernel ------------------------
__global__ void __launch_bounds__(TPB)
imgbuf_async_fanout(const vfloat4* __restrict__ inputs,
                    const vfloat4* __restrict__ buffer,
                    vfloat4* __restrict__ out) {
    __shared__ vfloat4 lds[CHUNK4];   // 16 KB

    const int j   = (int)blockIdx.y + 1;          // source frame index in combined[], 1..143
    const int c0  = (int)blockIdx.x * CHUNK4;     // chunk base within frame (float4 units)
    const int tid = (int)threadIdx.x;

    const vfloat4* src = (j < KBUF)
        ? (buffer + (size_t)j * FR4 + c0)
        : (inputs + (size_t)(j - KBUF) * FR4 + c0);

    // Async DMA: global -> LDS, 4 x b128 per thread (one chunk per block).
#pragma unroll
    for (int u = 0; u < F4_PER_THREAD; ++u) {
        const int p = tid + TPB * u;
        const unsigned loff = lds_byte_offset((void*)&lds[p]);
        asm volatile("global_load_async_to_lds_b128 %0, %1, off"
                     :
                     : "v"(loff), "v"(src + p)
                     : "memory");
    }

    // Wait for our wave's async loads to land in LDS. Each thread only
    // stores LDS lines it loaded itself -> no cross-wave barrier needed.
    wait_asynccnt0();

    // Fan out to every output window that contains frame j:
    //   k in [max(0, j-128), min(15, j-1)],  i = j-1-k
    const int klo = (j - BATCH) > 0 ? (j - BATCH) : 0;
    const int khi = (j - 1) < (KBUF - 1) ? (j - 1) : (KBUF - 1);

    for (int k = klo; k <= khi; ++k) {
        const int i = j - 1 - k;
        vfloat4* dst = out + (size_t)(i * KBUF + k) * FR4 + c0;
#pragma unroll
        for (int u = 0; u < F4_PER_THREAD; ++u) {
            const int p = tid + TPB * u;
            const unsigned loff = lds_byte_offset((void*)&lds[p]);
            // Non-temporal: 402 MB write-once stream, don't rinse L2.
            asm volatile("global_store_async_from_lds_b128 %0, %1, off th:TH_STORE_NT"
                         :
                         : "v"(dst + p), "v"(loff)
                         : "memory");
        }
    }

    wait_asynccnt0();   // all DMA stores complete before wave exit
}

// ---------------- direct register-copy fallback (not launched) -------------
// B128 loads (L2-temporal, 16x reuse) + non-temporal B128 stores.
__global__ void __launch_bounds__(TPB)
imgbuf_direct(const vfloat4* __restrict__ inputs,
              const vfloat4* __restrict__ buffer,
              vfloat4* __restrict__ out) {
    const int w = (int)blockIdx.y;        // i*16 + k, 0..2047
    const int i = w >> 4;
    const int k = w & 15;
    const int j = i + 1 + k;              // 1..143
    const vfloat4* src = (j < KBUF) ? (buffer + (size_t)j * FR4)
                                    : (inputs + (size_t)(j - KBUF) * FR4);
    vfloat4* dst = out + (size_t)w * FR4;

    int p = (int)blockIdx.x * (TPB * 4) + (int)threadIdx.x;
#pragma unroll
    for (int u = 0; u < 4; ++u, p += TPB) {
        const vfloat4 v = src[p];
        __builtin_nontemporal_store(v, dst + p);
    }
}

extern "C" void kernel_launch(void* const* d_in, const int* in_sizes, int n_in,
                              void* d_out, int out_size, void* d_ws, size_t ws_size,
                              hipStream_t stream) {
    (void)in_sizes; (void)n_in; (void)d_ws; (void)ws_size; (void)out_size;
    const vfloat4* inputs = (const vfloat4*)d_in[0];   // [128,128,128,3] f32
    const vfloat4* buffer = (const vfloat4*)d_in[1];   // [16,128,128,3]  f32
    vfloat4* out = (vfloat4*)d_out;                    // [128,16,128,128,3] f32

    // One block per (chunk, source-frame j); every output float4 written once.
    dim3 grid(NCHUNK, NSRC);   // (12, 143)
    imgbuf_async_fanout<<<grid, TPB, 0, stream>>>(inputs, buffer, out);
}